// MessagePassing_12841952215349
// MI455X (gfx1250) — compile-verified
//
#include <hip/hip_runtime.h>
#include <hip/hip_bf16.h>
#include <math.h>

// ---------------------------------------------------------------------------
// MessagePassing on MI455X (gfx1250, wave32).
// fp32 WMMA (V_WMMA_F32_16X16X4_F32) for all GEMMs; fused edge GEMM +
// message + scatter (segment_sum via GLOBAL_ATOMIC_ADD_F32).
// xd/xh (154 MB) staged in d_ws -> L2-resident (192 MB L2) for edge gathers.
// W_edge (96 KB) staged once per block in LDS (320 KB/WGP) -> 8x less L2
// weight traffic in the dominant edge kernel.
// ---------------------------------------------------------------------------

typedef float v2f __attribute__((ext_vector_type(2)));
typedef float v8f __attribute__((ext_vector_type(8)));

#define H_DIM 128
#define C3H   384          // 3*H
#define F_DIM 64

__constant__ const float INV_SQRT_3 = 0.57735026918962576f;
__constant__ const float INV_SQRT_H = 0.08838834764831845f;   // 1/sqrt(128)
__constant__ const float SILU_SCALE = 1.6666666666666667f;    // 1/0.6

// fp32 WMMA: D(16x16) = A(16x4) * B(4x16) + C
__device__ __forceinline__ v8f wmma4(v2f a, v2f b, v8f c) {
  return __builtin_amdgcn_wmma_f32_16x16x4_f32(
      /*neg_a=*/false, a, /*neg_b=*/false, b,
      /*c_mod=*/(short)0, c, /*reuse_a=*/false, /*reuse_b=*/false);
}

// A fragment (row-major A, leading dim lda):
// VGPR0: lanes0-15 K=kb+0, lanes16-31 K=kb+2 ; VGPR1: K=kb+1 / K=kb+3
// Contiguous K pair -> force a single 64-bit load.
__device__ __forceinline__ v2f load_a_frag(const float* A, int lda, int kb, int lane) {
  int m  = lane & 15;
  int k0 = kb + ((lane >> 4) << 1);
  return *(const v2f*)(A + (size_t)m * lda + k0);
}

// B fragment (row-major B [K x Ncols]): lane = column, rows mirror A's K split.
__device__ __forceinline__ v2f load_b_frag(const float* B, int ldb, int kb, int n0, int lane) {
  int n  = n0 + (lane & 15);
  int k0 = kb + ((lane >> 4) << 1);
  v2f b;
  b.x = B[k0 * ldb + n];
  b.y = B[(k0 + 1) * ldb + n];
  return b;
}

// ---------------------------------------------------------------------------
// Kernel 1: per 16-node tile (one tile per wave, 5 waves / block):
//   xd = x_defect @ W_defect + b_defect            [N,384]
//   h1 = ScaledSiLU(x @ W_x1 + b_x1)               (LDS staging)
//   xh = h1 @ W_x2 + b_x2                          [N,384]
// Weights are < 1 MB total -> L2-resident; cost of this kernel is ~20% of
// the edge phase, so structure kept simple.
// ---------------------------------------------------------------------------
__global__ __launch_bounds__(160) void node_gemm_kernel(
    const float* __restrict__ x, const float* __restrict__ x_defect,
    const float* __restrict__ W_defect, const float* __restrict__ b_defect,
    const float* __restrict__ W_x1,     const float* __restrict__ b_x1,
    const float* __restrict__ W_x2,     const float* __restrict__ b_x2,
    float* __restrict__ xd_out, float* __restrict__ xh_out, int ntiles)
{
  __shared__ float h1_lds[5][16 * 64];        // per-wave h1 tile (20 KB)

  const int lane = threadIdx.x & 31;
  const int wave = threadIdx.x >> 5;
  int tile = blockIdx.x * 5 + wave;
  if (tile >= ntiles) tile = ntiles - 1;      // idempotent stores -> clamp OK
  const int row0 = tile * 16;
  const int nlo  = lane & 15;
  const int mhi  = (lane >> 4) << 3;          // 0 or 8

  // ---- xd = x_defect @ W_defect + b_defect ----
  v2f a_xd[32];
#pragma unroll
  for (int k = 0; k < 32; ++k)
    a_xd[k] = load_a_frag(x_defect + (size_t)row0 * H_DIM, H_DIM, k * 4, lane);

  for (int ct = 0; ct < 24; ++ct) {
    v8f acc = {};
#pragma unroll
    for (int k = 0; k < 32; ++k)
      acc = wmma4(a_xd[k], load_b_frag(W_defect, C3H, k * 4, ct * 16, lane), acc);
    const int n = ct * 16 + nlo;
    const float bias = b_defect[n];
#pragma unroll
    for (int v = 0; v < 8; ++v)
      xd_out[(size_t)(row0 + mhi + v) * C3H + n] = acc[v] + bias;
  }

  // ---- h1 = ScaledSiLU(x @ W_x1 + b_x1) -> LDS ----
  v2f a_x[32];
#pragma unroll
  for (int k = 0; k < 32; ++k)
    a_x[k] = load_a_frag(x + (size_t)row0 * H_DIM, H_DIM, k * 4, lane);

  float* h1 = h1_lds[wave];
  for (int ct = 0; ct < 4; ++ct) {
    v8f acc = {};
#pragma unroll
    for (int k = 0; k < 32; ++k)
      acc = wmma4(a_x[k], load_b_frag(W_x1, 64, k * 4, ct * 16, lane), acc);
    const int n = ct * 16 + nlo;
    const float bias = b_x1[n];
#pragma unroll
    for (int v = 0; v < 8; ++v) {
      float t = acc[v] + bias;
      h1[(mhi + v) * 64 + n] = t / (1.0f + __expf(-t)) * SILU_SCALE;
    }
  }
  __syncthreads();

  // ---- xh = h1 @ W_x2 + b_x2 ----
  v2f a_h[16];
#pragma unroll
  for (int k = 0; k < 16; ++k)
    a_h[k] = load_a_frag(h1, 64, k * 4, lane);

  for (int ct = 0; ct < 24; ++ct) {
    v8f acc = {};
#pragma unroll
    for (int k = 0; k < 16; ++k)
      acc = wmma4(a_h[k], load_b_frag(W_x2, C3H, k * 4, ct * 16, lane), acc);
    const int n = ct * 16 + nlo;
    const float bias = b_x2[n];
#pragma unroll
    for (int v = 0; v < 8; ++v)
      xh_out[(size_t)(row0 + mhi + v) * C3H + n] = acc[v] + bias;
  }
}

// ---------------------------------------------------------------------------
// Kernel 2 (fused): per 16-edge tile (one per wave, 8 waves / block):
//   rbf = edge_feat(16x64) @ W_edge(64x384) + b_edge   (WMMA, B from LDS)
//   msg = xh[j]*(xd[j]+xd[i])*rbf/sqrt(3)
//   scatter-add thirds into d_vec / d_x with GLOBAL_ATOMIC_ADD_F32.
// W_edge (96 KB) + b_edge staged once per block in LDS: the 8 waves would
// otherwise each stream the full weight matrix from L2 (8x duplication).
// E/16 = 25000 tiles, 8 waves/block -> 3125 blocks exactly (no clamping;
// atomics must not be duplicated).
// ---------------------------------------------------------------------------
__global__ __launch_bounds__(256) void edge_fused_kernel(
    const float* __restrict__ edge_feat,   // [E,64]
    const float* __restrict__ edge_vector, // [E,3]
    const int*   __restrict__ edge_index,  // [2,E]
    const float* __restrict__ W_edge, const float* __restrict__ b_edge,
    const float* __restrict__ xd,          // [N,384]
    const float* __restrict__ xh,          // [N,384]
    const float* __restrict__ vec,         // [N,3,128]
    float* __restrict__ dx_out,            // [N,128]   (d_out, zeroed)
    float* __restrict__ dvec_out,          // [N,3,128] (d_out, zeroed)
    int E, int ntiles)
{
  __shared__ float sW[F_DIM * C3H];        // 96 KB: W_edge, row-major [64,384]
  __shared__ float sb[C3H];
  __shared__ int   jj[8][16];
  __shared__ int   ii[8][16];

  const int tid  = threadIdx.x;
  const int lane = tid & 31;
  const int wave = tid >> 5;
  const int tile = blockIdx.x * 8 + wave;
  const bool active = (tile < ntiles);
  const int e0 = (active ? tile : 0) * 16;

  // Block-cooperative staging of W_edge (b128 loads) and b_edge.
  {
    const float4* src = (const float4*)W_edge;
    float4*       dst = (float4*)sW;
#pragma unroll
    for (int i = 0; i < (F_DIM * C3H) / 4 / 256; ++i)   // 24 iters
      dst[tid + i * 256] = src[tid + i * 256];
    if (tid < C3H) sb[tid] = b_edge[tid];
    if (tid >= 256 - 0 && false) {}                     // (keep structure simple)
  }
  if (active && lane < 16) {
    jj[wave][lane] = edge_index[e0 + lane];       // j = edge_index[0]
    ii[wave][lane] = edge_index[E + e0 + lane];   // i = edge_index[1]
  }
  __syncthreads();

  if (!active) return;

  const int nlo = lane & 15;
  const int mhi = (lane >> 4) << 3;

  // Preload edge_feat A fragments (K = 64 -> 16 steps), 64-bit loads.
  v2f a_e[16];
#pragma unroll
  for (int k = 0; k < 16; ++k)
    a_e[k] = load_a_frag(edge_feat + (size_t)e0 * F_DIM, F_DIM, k * 4, lane);

  for (int ct = 0; ct < 24; ++ct) {
    v8f acc = {};
#pragma unroll
    for (int k = 0; k < 16; ++k)
      acc = wmma4(a_e[k], load_b_frag(sW, C3H, k * 4, ct * 16, lane), acc);

    const int   chan = ct * 16 + nlo;             // wave-uniform third
    const float bias = sb[chan];

#pragma unroll
    for (int v = 0; v < 8; ++v) {
      const int m = mhi + v;                      // edge within tile
      const int e = e0 + m;
      const int j = jj[wave][m];
      const int i = ii[wave][m];
      const float rbf = acc[v] + bias;
      const float msg = xh[(size_t)j * C3H + chan] *
                        (xd[(size_t)j * C3H + chan] + xd[(size_t)i * C3H + chan]) *
                        rbf * INV_SQRT_3;

      if (chan < H_DIM) {                         // x_ji1 * vec[j]
        const int h = chan;
#pragma unroll
        for (int d = 0; d < 3; ++d)
          atomicAdd(&dvec_out[((size_t)i * 3 + d) * H_DIM + h],
                    msg * vec[((size_t)j * 3 + d) * H_DIM + h] * INV_SQRT_H);
      } else if (chan < 2 * H_DIM) {              // x_ji2 * edge_vector
        const int h = chan - H_DIM;
#pragma unroll
        for (int d = 0; d < 3; ++d)
          atomicAdd(&dvec_out[((size_t)i * 3 + d) * H_DIM + h],
                    msg * edge_vector[(size_t)e * 3 + d] * INV_SQRT_H);
      } else {                                    // x_ji3 -> d_x
        atomicAdd(&dx_out[(size_t)i * H_DIM + (chan - 2 * H_DIM)], msg);
      }
    }
  }
}

// ---------------------------------------------------------------------------
// Kernel 3: vector activation, in place on d_vec region.
//   gvec = v @ W_gvec ; lvec = v @ W_lvec ; gated blend per channel.
// W_lvec (64 KB) staged in LDS; 40 nodes per block.
// ---------------------------------------------------------------------------
#define NODES_PER_BLOCK 40
__global__ __launch_bounds__(256) void vec_act_kernel(
    const float* __restrict__ W_lvec,  // [128,128]
    const float* __restrict__ W_gvec,  // [128,1]
    float* __restrict__ dvec,          // [N,3,128] in/out
    int N)
{
  __shared__ float sW[H_DIM * H_DIM];  // 64 KB
  __shared__ float sg[H_DIM];
  __shared__ float sv[2][3 * H_DIM];

  const int tid = threadIdx.x;
  for (int idx = tid; idx < H_DIM * H_DIM; idx += 256) sW[idx] = W_lvec[idx];
  if (tid < H_DIM) sg[tid] = W_gvec[tid];

  const int g = tid >> 7;        // node group (0/1)
  const int h = tid & 127;       // channel
  const int base = blockIdx.x * NODES_PER_BLOCK;

  for (int it = 0; it < NODES_PER_BLOCK / 2; ++it) {
    const int n = base + it * 2 + g;
    __syncthreads();             // sW ready (1st iter) / sv reuse safe
    if (n < N) {
#pragma unroll
      for (int d = 0; d < 3; ++d)
        sv[g][d * H_DIM + h] = dvec[((size_t)n * 3 + d) * H_DIM + h];
    }
    __syncthreads();
    if (n < N) {
      float lv[3] = {0.f, 0.f, 0.f};
      float gv[3] = {0.f, 0.f, 0.f};
#pragma unroll 4
      for (int k = 0; k < H_DIM; ++k) {
        const float w = sW[k * H_DIM + h];
        const float s = sg[k];
#pragma unroll
        for (int d = 0; d < 3; ++d) {
          const float vk = sv[g][d * H_DIM + k];
          lv[d] = fmaf(vk, w, lv[d]);
          gv[d] = fmaf(vk, s, gv[d]);
        }
      }
      const float dot = lv[0] * gv[0] + lv[1] * gv[1] + lv[2] * gv[2];
      const bool pass = (dot >= 0.0f);
#pragma unroll
      for (int d = 0; d < 3; ++d) {
        const float out = pass ? lv[d] : (lv[d] + gv[d]) * 0.5f;
        dvec[((size_t)n * 3 + d) * H_DIM + h] = out;
      }
    }
  }
}

// ---------------------------------------------------------------------------
extern "C" void kernel_launch(void* const* d_in, const int* in_sizes, int n_in,
                              void* d_out, int out_size, void* d_ws, size_t ws_size,
                              hipStream_t stream) {
  const float* x           = (const float*)d_in[0];
  const float* x_defect    = (const float*)d_in[1];
  const float* vec         = (const float*)d_in[2];
  const float* edge_feat   = (const float*)d_in[3];
  const float* edge_vector = (const float*)d_in[4];
  const int*   edge_index  = (const int*)  d_in[5];
  const float* W_defect    = (const float*)d_in[6];
  const float* b_defect    = (const float*)d_in[7];
  const float* W_x1        = (const float*)d_in[8];
  const float* b_x1        = (const float*)d_in[9];
  const float* W_x2        = (const float*)d_in[10];
  const float* b_x2        = (const float*)d_in[11];
  const float* W_edge      = (const float*)d_in[12];
  const float* b_edge      = (const float*)d_in[13];
  const float* W_lvec      = (const float*)d_in[14];
  const float* W_gvec      = (const float*)d_in[15];

  const int N = in_sizes[0] / H_DIM;   // 50000
  const int E = in_sizes[3] / F_DIM;   // 400000

  float* xd = (float*)d_ws;                       // [N,384]
  float* xh = xd + (size_t)N * C3H;               // [N,384]  (together ~154 MB)
  float* dx_out   = (float*)d_out;                // [N,128]
  float* dvec_out = dx_out + (size_t)N * H_DIM;   // [N,3,128]

  // Zero the accumulation targets (segment_sum) every call.
  hipMemsetAsync(d_out, 0, (size_t)out_size * sizeof(float), stream);

  // 1) Node GEMMs (WMMA fp32), one 16-node tile per wave, 5 waves/block.
  const int ntilesN = (N + 15) / 16;              // 3125
  const int blocks1 = (ntilesN + 4) / 5;          // 625
  node_gemm_kernel<<<blocks1, 160, 0, stream>>>(
      x, x_defect, W_defect, b_defect, W_x1, b_x1, W_x2, b_x2, xd, xh, ntilesN);

  // 2) Fused edge GEMM + message + scatter, 8 waves/block, W_edge in LDS.
  const int ntilesE = (E + 15) / 16;              // 25000
  const int blocks2 = (ntilesE + 7) / 8;          // 3125
  edge_fused_kernel<<<blocks2, 256, 0, stream>>>(
      edge_feat, edge_vector, edge_index, W_edge, b_edge,
      xd, xh, vec, dx_out, dvec_out, E, ntilesE);

  // 3) Vector activation in place on d_vec region.
  const int blocks3 = (N + NODES_PER_BLOCK - 1) / NODES_PER_BLOCK;  // 1250
  vec_act_kernel<<<blocks3, 256, 0, stream>>>(W_lvec, W_gvec, dvec_out, N);
}